// AMSELoss_55104430407745
// MI455X (gfx1250) — compile-verified
//
#include <hip/hip_runtime.h>

#define NLAT 361
#define NLON 720
#define KDEG 360          // reference drops last degree: C[:,:,:-1,:]
#define NCOL 64           // 16 (b*c) x 2 (P/T) x 2 (re/im)
#define EPSF 1e-7f
#define TWO_PI_F 6.2831853071795864769f

typedef __attribute__((ext_vector_type(2))) float v2f;
typedef __attribute__((ext_vector_type(8))) float v8f;

// ---------------------------------------------------------------------------
// Kernel 1: naive real DFT (rfft) along longitude for one latitude row / one
// tensor. Row data + twiddle tables staged in LDS; each thread owns one
// (m, batch*chan) output bin. F scaled by 2*pi/NLON like the reference.
// Output layout: Fbuf[m][j][col], col = bc*4 + tensor*2 + {re,im}.
// ---------------------------------------------------------------------------
__global__ __launch_bounds__(256) void dft_kernel(
    const float* __restrict__ pred, const float* __restrict__ targ,
    float* __restrict__ Fbuf)
{
  __shared__ float rows[16 * 721];   // 721 stride: conflict-free (gcd(17,64)=1)
  __shared__ float ct[NLON];
  __shared__ float st[NLON];

  const int j   = blockIdx.x;        // latitude
  const int mc  = blockIdx.y;        // m-chunk of 16
  const int t   = blockIdx.z;        // 0 = prediction, 1 = target
  const int tid = threadIdx.x;
  const float* __restrict__ src = (t == 0) ? pred : targ;

  for (int i = tid; i < 16 * NLON; i += 256) {
    int row = i / NLON, n = i - row * NLON;
    rows[row * 721 + n] = src[((size_t)row * NLAT + j) * NLON + n];
  }
  for (int i = tid; i < NLON; i += 256) {
    float ang = (TWO_PI_F / (float)NLON) * (float)i;
    float s, c;
    __sincosf(ang, &s, &c);
    ct[i] = c; st[i] = s;
  }
  __syncthreads();

  const int bc   = tid & 15;
  const int msub = tid >> 4;
  const int m    = mc * 16 + msub;
  if (m < NLAT) {
    const float* __restrict__ r = &rows[bc * 721];
    float re = 0.f, im = 0.f;
    int idx = 0;                     // (m*n) mod NLON, updated incrementally
    for (int n = 0; n < NLON; ++n) {
      float v = r[n];
      re = fmaf(v,  ct[idx], re);
      im = fmaf(-v, st[idx], im);    // rfft: e^{-i 2pi mn/N}
      idx += m; if (idx >= NLON) idx -= NLON;
    }
    const float scale = TWO_PI_F / (float)NLON;
    size_t base = ((size_t)m * NLAT + j) * NCOL + (size_t)bc * 4 + (size_t)t * 2;
    Fbuf[base + 0] = re * scale;
    Fbuf[base + 1] = im * scale;
  }
}

// ---------------------------------------------------------------------------
// Kernel 2: Legendre transform as 361 independent GEMMs using f32 WMMA.
// For each m:  C[l][n] = sum_j (leg[l][m][j] * w[j]) * Fbuf[m][j][n]
// M = 361 degrees, K = 361 lats, N = 64 columns. One block = (m, 128-degree
// supertile); 8 waves, each wave owns a 16-degree tile and 4 N-tiles.
// WMMA shape: D(16x16) += A(16x4) x B(4x16), f32 everywhere.
// B is staged pair-major in LDS (float2 over adjacent k) so each WMMA B
// operand is a single aligned ds_load_b64 into an even VGPR pair (no repack).
// ---------------------------------------------------------------------------
__global__ __launch_bounds__(256) void legendre_wmma_kernel(
    const float* __restrict__ leg, const float* __restrict__ w,
    const float* __restrict__ Fbuf, float* __restrict__ Cbuf)
{
  __shared__ float As[128 * 33];     // [l_local][k_local], padded stride
  __shared__ v2f   Bs2[16 * 65];     // [k_local/2][n] -> (B[k][n], B[k+1][n])

  const int m    = blockIdx.x;
  const int l0   = blockIdx.y * 128;
  const int tid  = threadIdx.x;
  const int lane = tid & 31;
  const int wave = tid >> 5;
  const int lrow  = lane & 15;        // row within tile / N index
  const int khalf = (lane >> 4) << 1; // lanes 0-15 -> K 0,1 ; 16-31 -> K 2,3

  v8f acc[4] = {};

  for (int k0 = 0; k0 < NLAT; k0 += 32) {
    __syncthreads();
    // Stage A = leg[l][m][k] * w[k] (coalesced over k, zero-padded).
    for (int i = tid; i < 128 * 32; i += 256) {
      int ll = i >> 5, kk = i & 31;
      int l = l0 + ll, k = k0 + kk;
      float a = 0.f;
      if (l < NLAT && k < NLAT)
        a = leg[((size_t)l * NLAT + m) * NLAT + k] * w[k];
      As[ll * 33 + kk] = a;
    }
    // Stage B = Fbuf[m][k][n], pair-major over k (fully coalesced rows of 64).
    for (int i = tid; i < 32 * 64; i += 256) {
      int kk = i >> 6, n = i & 63;
      int k = k0 + kk;
      float v = (k < NLAT) ? Fbuf[((size_t)m * NLAT + k) * NCOL + n] : 0.f;
      Bs2[(kk >> 1) * 65 + n][kk & 1] = v;
    }
    // Prefetch next chunk of the leg stream into cache.
    if (k0 + 32 < NLAT)
      __builtin_prefetch(&leg[((size_t)(l0 + (tid >> 1)) * NLAT + m) * NLAT + k0 + 32], 0, 1);
    __syncthreads();

    const int la = wave * 16 + lrow;  // this wave's A row in LDS
    #pragma unroll
    for (int ks = 0; ks < 32; ks += 4) {
      v2f a;
      a[0] = As[la * 33 + ks + khalf + 0];
      a[1] = As[la * 33 + ks + khalf + 1];
      const int prow = (ks + khalf) >> 1;   // pair row: holds (k, k+1)
      #pragma unroll
      for (int nt = 0; nt < 4; ++nt) {
        v2f b = Bs2[prow * 65 + nt * 16 + lrow];
        // 8 args: (neg_a, A, neg_b, B, c_mod, C, reuse_a, reuse_b)
        acc[nt] = __builtin_amdgcn_wmma_f32_16x16x4_f32(
            false, a, false, b, (short)0, acc[nt], false, false);
      }
    }
  }

  // Store D: VGPR r -> M = r (lanes 0-15) / r+8 (lanes 16-31), N = lrow.
  const int moff = (lane >> 4) << 3;
  #pragma unroll
  for (int r = 0; r < 8; ++r) {
    int l = l0 + wave * 16 + r + moff;
    if (l < NLAT) {
      #pragma unroll
      for (int nt = 0; nt < 4; ++nt)
        Cbuf[((size_t)m * NLAT + l) * NCOL + nt * 16 + lrow] = acc[nt][r];
    }
  }
}

// ---------------------------------------------------------------------------
// Kernel 3: per (bc, degree l) triangular reduction over m (m<=l), with the
// reference's m=0 weight-1 / m>0 weight-2 doubling, producing
// amplitude + decorrelation per degree.
// ---------------------------------------------------------------------------
__global__ __launch_bounds__(128) void psd_kernel(
    const float* __restrict__ Cbuf, float* __restrict__ perdeg)
{
  __shared__ float red[4][128];
  const int l   = blockIdx.x;        // 0..359
  const int bc  = blockIdx.y;        // 0..15
  const int tid = threadIdx.x;

  float pp = 0.f, tp = 0.f, sr = 0.f, si = 0.f;
  for (int m = tid; m <= l; m += 128) {
    const float* cp = Cbuf + ((size_t)m * NLAT + l) * NCOL + (size_t)bc * 4;
    float pr = cp[0], pi = cp[1], tr = cp[2], ti = cp[3];
    float wc = (m == 0) ? 1.f : 2.f;
    pp = fmaf(wc, fmaf(pr, pr, pi * pi), pp);
    tp = fmaf(wc, fmaf(tr, tr, ti * ti), tp);
    sr = fmaf(wc, fmaf(pr, tr, pi * ti), sr);
    si = fmaf(wc, fmaf(pr, ti, -pi * tr), si);
  }
  red[0][tid] = pp; red[1][tid] = tp; red[2][tid] = sr; red[3][tid] = si;
  __syncthreads();
  for (int s = 64; s > 0; s >>= 1) {
    if (tid < s) {
      #pragma unroll
      for (int q = 0; q < 4; ++q) red[q][tid] += red[q][tid + s];
    }
    __syncthreads();
  }
  if (tid == 0) {
    pp = red[0][0] + EPSF;           // _psd adds EPS
    tp = red[1][0] + EPSF;
    sr = red[2][0]; si = red[3][0];
    float mag   = sqrtf(sr * sr + si * si);
    float denom = sqrtf(pp * tp + EPSF);
    float coh   = mag / (denom + EPSF);
    coh = fminf(fmaxf(coh, 0.f), 1.f);
    float dsq = sqrtf(pp) - sqrtf(tp);
    float amplitude = dsq * dsq;
    float decor = 2.f * fmaxf(pp, tp) * (1.f - coh);
    perdeg[bc * KDEG + l] = amplitude + decor;
  }
}

// ---------------------------------------------------------------------------
// Kernel 4: deterministic weighted mean -> scalar loss with NaN guard.
// loss = sum_{bc,l} perdeg[bc][l] * weights[c] / (360 * 16)
// ---------------------------------------------------------------------------
__global__ __launch_bounds__(256) void loss_kernel(
    const float* __restrict__ perdeg, const float* __restrict__ weights,
    float* __restrict__ out)
{
  __shared__ float red[256];
  const int tid = threadIdx.x;
  float acc = 0.f;
  for (int i = tid; i < 16 * KDEG; i += 256) {
    int bc = i / KDEG;
    acc = fmaf(perdeg[i], weights[bc & 7], acc);
  }
  red[tid] = acc;
  __syncthreads();
  for (int s = 128; s > 0; s >>= 1) {
    if (tid < s) red[tid] += red[tid + s];
    __syncthreads();
  }
  if (tid == 0) {
    float loss = red[0] / (float)(KDEG * 16);
    out[0] = (loss != loss) ? 1000000.0f : loss;
  }
}

// ---------------------------------------------------------------------------
extern "C" void kernel_launch(void* const* d_in, const int* in_sizes, int n_in,
                              void* d_out, int out_size, void* d_ws, size_t ws_size,
                              hipStream_t stream) {
  (void)in_sizes; (void)n_in; (void)out_size; (void)ws_size;
  const float* pred    = (const float*)d_in[0];
  const float* targ    = (const float*)d_in[1];
  const float* weights = (const float*)d_in[2];
  const float* leg     = (const float*)d_in[3];
  const float* w       = (const float*)d_in[4];
  float* out = (float*)d_out;

  const size_t tile = (size_t)NLAT * NLAT * NCOL;   // 8,340,544 floats
  float* Fbuf   = (float*)d_ws;                     // [m][j][64]
  float* Cbuf   = Fbuf + tile;                      // [m][l][64]
  float* perdeg = Cbuf + tile;                      // [bc][360]

  dft_kernel<<<dim3(NLAT, 23, 2), 256, 0, stream>>>(pred, targ, Fbuf);
  legendre_wmma_kernel<<<dim3(NLAT, 3), 256, 0, stream>>>(leg, w, Fbuf, Cbuf);
  psd_kernel<<<dim3(KDEG, 16), 128, 0, stream>>>(Cbuf, perdeg);
  loss_kernel<<<1, 256, 0, stream>>>(perdeg, weights, out);
}